// HybridSpikeEncoder_90108413870272
// MI455X (gfx1250) — compile-verified
//
#include <hip/hip_runtime.h>
#include <hip/hip_bf16.h>
#include <math.h>

// HybridSpikeEncoder forward for MI455X (gfx1250).
// Memory-bound: 32 MB read + 512 MB NT-stored output (~23 us @ 23.3 TB/s floor).
// - gfx1250 async global->LDS loads (ASYNCcnt) double-buffer the z staging
// - ds_load_b128 read-back, global_store_b128 th:NT output stream (4 cols/thread)

#define TSTEPS 16
#define BLK    256   // threads per block (8 wave32s); covers BLK*4 columns
#define ROWS   8     // batch rows per block

typedef float fvec4 __attribute__((ext_vector_type(4)));
typedef __attribute__((address_space(1))) int* as1_i32p;  // global int*
typedef __attribute__((address_space(3))) int* as3_i32p;  // LDS int*

#if __has_builtin(__builtin_amdgcn_global_load_async_to_lds_b32)
#define HAVE_ASYNC_BUILTIN 1
#endif

// Async copy of 4 consecutive DWORDs: the instruction's immediate offset is
// added to BOTH the LDS and global addresses (CDNA5 ISA), so a
// single address pair feeds 4 issues at offsets 0/4/8/12.
__device__ __forceinline__ void async_g2l_b128_as_4xb32(const float* g, float* l) {
#ifdef HAVE_ASYNC_BUILTIN
  as1_i32p gp = (as1_i32p)(unsigned long long)g;
  as3_i32p lp = (as3_i32p)(unsigned)(unsigned long long)l;
  __builtin_amdgcn_global_load_async_to_lds_b32(gp, lp, 0,  0);
  __builtin_amdgcn_global_load_async_to_lds_b32(gp, lp, 4,  0);
  __builtin_amdgcn_global_load_async_to_lds_b32(gp, lp, 8,  0);
  __builtin_amdgcn_global_load_async_to_lds_b32(gp, lp, 12, 0);
#else
  asm volatile("global_load_async_to_lds_b32 %0, %1, off\n\t"
               "global_load_async_to_lds_b32 %0, %1, off offset:4\n\t"
               "global_load_async_to_lds_b32 %0, %1, off offset:8\n\t"
               "global_load_async_to_lds_b32 %0, %1, off offset:12"
               :
               : "v"((unsigned)(unsigned long long)l),
                 "v"((unsigned long long)g)
               : "memory");
#endif
}

template <int N>
__device__ __forceinline__ void wait_asynccnt() {
#if __has_builtin(__builtin_amdgcn_s_wait_asynccnt)
  __builtin_amdgcn_s_wait_asynccnt(N);
  asm volatile("" ::: "memory");   // keep LDS reads below the wait
#else
  asm volatile("s_wait_asynccnt %0" :: "i"(N) : "memory");
#endif
}

__global__ void __launch_bounds__(BLK)
spike_encoder_kernel(const float* __restrict__ z,
                     const float* __restrict__ beta_raw,
                     const float* __restrict__ thresh_param,
                     float* __restrict__ out,
                     int B, int D) {
  __shared__ float zbuf[2][BLK * 4];   // double-buffered z staging (8 KB)

  const int tid = threadIdx.x;
  const int j0  = blockIdx.x * (BLK * 4) + tid * 4;  // first of 4 owned columns
  const int b0  = blockIdx.y * ROWS;

  int nrows = B - b0;
  if (nrows <= 0) return;
  if (nrows > ROWS) nrows = ROWS;

  const bool vec_ok = (j0 + 3 < D);   // full float4 lane (always true for D=1024)

  if (vec_ok) {
    // ---- per-column parameters, amortized over ROWS rows ----
    fvec4 beta, omb, th, jf;
    #pragma unroll
    for (int c = 0; c < 4; ++c) {
      const float braw = beta_raw[j0 + c];
      beta[c] = 1.0f / (1.0f + __expf(-braw));    // sigmoid
      omb[c]  = 1.0f - beta[c];
      th[c]   = fabsf(thresh_param[j0 + c]);
      jf[c]   = (float)(j0 + c);
    }

    fvec4 phase[TSTEPS];
    #pragma unroll
    for (int t = 0; t < TSTEPS; ++t) {
      const float w = (6.28318530717958647692f * (float)t) / (float)TSTEPS;
      #pragma unroll
      for (int c = 0; c < 4; ++c) phase[t][c] = __sinf(w * jf[c]);
    }

    // ---- prologue: async-stage row b0 into buffer 0 ----
    async_g2l_b128_as_4xb32(z + (long long)b0 * D + j0, &zbuf[0][tid * 4]);

    for (int r = 0; r < nrows; ++r) {
      const int b   = b0 + r;
      const int cur = r & 1;

      if (r + 1 < nrows) {
        // Kick next row, then wait for current row's 4 issues only
        // (per-wave async loads complete in order: cnt<=4 => first 4 done).
        async_g2l_b128_as_4xb32(z + (long long)(b + 1) * D + j0,
                                &zbuf[cur ^ 1][tid * 4]);
        wait_asynccnt<4>();
      } else {
        wait_asynccnt<0>();
      }

      const fvec4 zv   = *(const fvec4*)&zbuf[cur][tid * 4];   // ds_load_b128
      const fvec4 rate = beta * zv;
      const fvec4 tb   = omb * zv;
      fvec4 mem = (fvec4)0.0f;

      float* orow = out + ((long long)b * TSTEPS) * (long long)D + j0;
      #pragma unroll
      for (int t = 0; t < TSTEPS; ++t) {
        fvec4 s;
        #pragma unroll
        for (int c = 0; c < 4; ++c) {
          mem[c] = mem[c] + rate[c] + tb[c] * phase[t][c];     // integrate
          s[c]   = (mem[c] - th[c] >= 0.0f) ? 1.0f : 0.0f;     // spike_fn
          mem[c] = mem[c] * (1.0f - s[c]);                     // reset
        }
        // 512 MB streaming output: b128 non-temporal, don't pollute L2.
        __builtin_nontemporal_store(s, (fvec4*)(orow + (long long)t * D));
      }
    }
  } else {
    // ---- scalar tail path (dead for D=1024): direct global loads ----
    #pragma unroll 1
    for (int c = 0; c < 4; ++c) {
      const int j = j0 + c;
      if (j >= D) break;
      const float beta = 1.0f / (1.0f + __expf(-beta_raw[j]));
      const float omb  = 1.0f - beta;
      const float th   = fabsf(thresh_param[j]);
      float phase[TSTEPS];
      #pragma unroll
      for (int t = 0; t < TSTEPS; ++t) {
        const float w = (6.28318530717958647692f * (float)t) / (float)TSTEPS;
        phase[t] = __sinf(w * (float)j);
      }
      for (int r = 0; r < nrows; ++r) {
        const int b = b0 + r;
        const float zv = z[(long long)b * D + j];
        const float rate = beta * zv, tb = omb * zv;
        float mem = 0.0f;
        float* orow = out + ((long long)b * TSTEPS) * (long long)D + j;
        #pragma unroll
        for (int t = 0; t < TSTEPS; ++t) {
          mem = mem + rate + tb * phase[t];
          const float s = (mem - th >= 0.0f) ? 1.0f : 0.0f;
          mem = mem * (1.0f - s);
          __builtin_nontemporal_store(s, orow + (long long)t * D);
        }
      }
    }
  }
}

extern "C" void kernel_launch(void* const* d_in, const int* in_sizes, int n_in,
                              void* d_out, int out_size, void* d_ws, size_t ws_size,
                              hipStream_t stream) {
  const float* z        = (const float*)d_in[0];  // (B, D) f32
  const float* beta_raw = (const float*)d_in[1];  // (D,)   f32
  const float* thr      = (const float*)d_in[2];  // (D,)   f32
  float*       out      = (float*)d_out;          // (B, T, D) f32

  const int D = in_sizes[1];
  const int B = (D > 0) ? in_sizes[0] / D : 0;

  dim3 grid((D + BLK * 4 - 1) / (BLK * 4), (B + ROWS - 1) / ROWS, 1);
  dim3 block(BLK, 1, 1);
  hipLaunchKernelGGL(spike_encoder_kernel, grid, block, 0, stream,
                     z, beta_raw, thr, out, B, D);
}